// GIN_26774826123547
// MI455X (gfx1250) — compile-verified
//
#include <hip/hip_runtime.h>
#include <hip/hip_bf16.h>

typedef float v2f __attribute__((ext_vector_type(2)));
typedef float v8f __attribute__((ext_vector_type(8)));

#define N_NODES   100000
#define N_EDGES   1200000
#define HDIM      64
#define N_GRAPHS  512
#define BN_EPS    1e-5f
#define NODE_F    (N_NODES * HDIM)      // 6,400,000
#define EDGE_F    (N_EDGES * HDIM)      // 76,800,000

// ---------------------------------------------------------------- utilities
__global__ void k_zero(float* __restrict__ p, int n) {
    int i = blockIdx.x * blockDim.x + threadIdx.x;
    if (i < n) p[i] = 0.f;
}

__global__ void k_copy(const float* __restrict__ src, float* __restrict__ dst, int n) {
    int i = blockIdx.x * blockDim.x + threadIdx.x;
    if (i < n) dst[i] = src[i];
}

// ------------------------------------------------- edge scatter (dominant cost)
// one thread per (edge, feature): threads 0..63 of a group share an edge ->
// coalesced 256B gather from h[src] and contiguous f32 atomics into sum[dst].
__global__ void k_edge_scatter(const float* __restrict__ h,
                               const int* __restrict__ src,
                               const int* __restrict__ dst,
                               float* __restrict__ sum) {
    int idx = blockIdx.x * blockDim.x + threadIdx.x;
    if (idx >= EDGE_F) return;
    int e = idx >> 6;
    int f = idx & 63;
    int s = src[e];
    int d = dst[e];
    atomicAdd(&sum[d * HDIM + f], h[s * HDIM + f]);
}

// ------------------------------------------------- 64-wide GEMM via f32 WMMA
// Y = act( X[nrows,64] @ W[64,64] + bias ), optional fused eval-mode BatchNorm.
// Block = 128 threads = 4 waves; each wave computes a 16-row x 64-col stripe
// with V_WMMA_F32_16X16X4_F32 (K=64 -> 16 steps x 4 N-tiles = 64 WMMAs/wave).
__global__ __launch_bounds__(128) void k_gemm64(
    const float* __restrict__ X, const float* __restrict__ W,
    const float* __restrict__ bias,
    const float* __restrict__ gamma, const float* __restrict__ beta,
    const float* __restrict__ rm,    const float* __restrict__ rv,
    float* __restrict__ Y, int nrows, int use_bn)
{
    const int lane = threadIdx.x & 31;
    const int wave = threadIdx.x >> 5;
    const int hi   = lane >> 4;          // 0 for lanes 0-15, 1 for lanes 16-31
    const int l16  = lane & 15;
    const int row0 = blockIdx.x * 64 + wave * 16;

    // A-fragment source row for this lane (clamped; zero-masked if OOB)
    const int   arow = row0 + l16;
    const int   crow = arow < (nrows - 1) ? arow : (nrows - 1);
    const float amsk = (arow < nrows) ? 1.f : 0.f;
    const float* xr  = X + crow * HDIM;

    v8f acc0 = {}, acc1 = {}, acc2 = {}, acc3 = {};

    #pragma unroll
    for (int s = 0; s < 16; ++s) {
        const int kk = 4 * s + 2 * hi;   // K index pair held by this lane
        v2f a;
        a.x = xr[kk]     * amsk;
        a.y = xr[kk + 1] * amsk;
        const float* w0 = W + kk * HDIM;         // row kk of W (64 cols)
        const float* w1 = W + (kk + 1) * HDIM;   // row kk+1
        {   v2f b; b.x = w0[l16];      b.y = w1[l16];
            acc0 = __builtin_amdgcn_wmma_f32_16x16x4_f32(false, a, false, b, (short)0, acc0, false, false); }
        {   v2f b; b.x = w0[16 + l16]; b.y = w1[16 + l16];
            acc1 = __builtin_amdgcn_wmma_f32_16x16x4_f32(false, a, false, b, (short)0, acc1, false, false); }
        {   v2f b; b.x = w0[32 + l16]; b.y = w1[32 + l16];
            acc2 = __builtin_amdgcn_wmma_f32_16x16x4_f32(false, a, false, b, (short)0, acc2, false, false); }
        {   v2f b; b.x = w0[48 + l16]; b.y = w1[48 + l16];
            acc3 = __builtin_amdgcn_wmma_f32_16x16x4_f32(false, a, false, b, (short)0, acc3, false, false); }
    }

    // Epilogue: y = relu(acc*sc + sh); plain linear: sc=1, sh=bias;
    // BN folded: sc = gamma*rsqrt(rv+eps), sh = (bias-rm)*sc + beta.
    #pragma unroll
    for (int t = 0; t < 4; ++t) {
        v8f acc = (t == 0) ? acc0 : (t == 1) ? acc1 : (t == 2) ? acc2 : acc3;
        const int col = 16 * t + l16;
        float b  = bias[col];
        float sc = 1.f, sh = b;
        if (use_bn) {
            float s_ = gamma[col] * rsqrtf(rv[col] + BN_EPS);
            sc = s_;
            sh = (b - rm[col]) * s_ + beta[col];
        }
        #pragma unroll
        for (int r = 0; r < 8; ++r) {
            const int row = row0 + r + hi * 8;   // C/D layout: M = r + 8*(lane>=16)
            if (row < nrows) {
                float v = acc[r] * sc + sh;
                Y[row * HDIM + col] = fmaxf(v, 0.f);
            }
        }
    }
}

// ------------------------------------------------- global add pool (per layer)
__global__ void k_pool(const float* __restrict__ h, const int* __restrict__ batch,
                       float* __restrict__ p, int off) {
    int idx = blockIdx.x * blockDim.x + threadIdx.x;
    if (idx >= NODE_F) return;
    int node = idx >> 6;
    int f    = idx & 63;
    atomicAdd(&p[batch[node] * 192 + off + f], h[idx]);
}

// ------------------------------------------------- classifier head
__global__ void k_head1(const float* __restrict__ p, const float* __restrict__ W,
                        const float* __restrict__ b, float* __restrict__ hid) {
    int idx = blockIdx.x * blockDim.x + threadIdx.x;   // 512*192
    if (idx >= N_GRAPHS * 192) return;
    int g = idx / 192, j = idx - g * 192;
    const float* pr = p + g * 192;
    float acc = b[j];
    #pragma unroll 4
    for (int k = 0; k < 192; ++k) acc += pr[k] * W[k * 192 + j];
    hid[idx] = fmaxf(acc, 0.f);
}

__global__ void k_head2(const float* __restrict__ hid, const float* __restrict__ W,
                        const float* __restrict__ b, float* __restrict__ out) {
    int g = blockIdx.x * blockDim.x + threadIdx.x;
    if (g >= N_GRAPHS) return;
    const float* hr = hid + g * 192;
    float c0 = b[0], c1 = b[1];
    #pragma unroll 4
    for (int k = 0; k < 192; ++k) {
        float h = hr[k];
        c0 += h * W[k * 2 + 0];
        c1 += h * W[k * 2 + 1];
    }
    out[g * 2 + 0] = c0;
    out[g * 2 + 1] = c1;
    float m  = fmaxf(c0, c1);
    float e0 = __expf(c0 - m), e1 = __expf(c1 - m);
    float inv = 1.f / (e0 + e1);
    out[N_GRAPHS * 2 + g * 2 + 0] = e0 * inv;
    out[N_GRAPHS * 2 + g * 2 + 1] = e1 * inv;
}

// ---------------------------------------------------------------- launcher
extern "C" void kernel_launch(void* const* d_in, const int* in_sizes, int n_in,
                              void* d_out, int out_size, void* d_ws, size_t ws_size,
                              hipStream_t stream) {
    (void)in_sizes; (void)n_in; (void)out_size; (void)ws_size;
    const float* x    = (const float*)d_in[0];
    const int*   edge = (const int*)d_in[1];
    const int*   src  = edge;                // edge_index[0]
    const int*   dst  = edge + N_EDGES;      // edge_index[1]
    const int*   batch = (const int*)d_in[2];
    auto P = [&](int i) { return (const float*)d_in[i]; };

    float* ws    = (float*)d_ws;
    float* B0    = ws;                        // sum buffer     (6.4M f32)
    float* B1    = B0 + (size_t)NODE_F;       // mid buffer     (6.4M f32)
    float* B2    = B1 + (size_t)NODE_F;       // layer output   (6.4M f32)
    float* pools = B2 + (size_t)NODE_F;       // 512 x 192
    float* hid   = pools + (size_t)N_GRAPHS * 192;   // 512 x 192

    const int TB = 256;
    const int gCopy    = (NODE_F + TB - 1) / TB;
    const int gScatter = (EDGE_F + TB - 1) / TB;
    const int gGemm    = (N_NODES + 63) / 64;
    const int gPoolZ   = (N_GRAPHS * 192 + TB - 1) / TB;

    k_zero<<<gPoolZ, TB, 0, stream>>>(pools, N_GRAPHS * 192);

    // One GIN layer: sum = hin + scatter(hin); mid = BN-MLP1(sum); h = MLP2(mid); pool.
    auto layer = [&](const float* hin, int poff, int pi) {
        k_copy<<<gCopy, TB, 0, stream>>>(hin, B0, NODE_F);
        k_edge_scatter<<<gScatter, TB, 0, stream>>>(hin, src, dst, B0);
        k_gemm64<<<gGemm, 128, 0, stream>>>(B0, P(pi + 0), P(pi + 1), P(pi + 2),
                                            P(pi + 3), P(pi + 4), P(pi + 5),
                                            B1, N_NODES, 1);
        k_gemm64<<<gGemm, 128, 0, stream>>>(B1, P(pi + 6), P(pi + 7),
                                            nullptr, nullptr, nullptr, nullptr,
                                            B2, N_NODES, 0);
        k_pool<<<gCopy, TB, 0, stream>>>(B2, batch, pools, poff);
    };

    layer(x,  0,   3);    // c1_*  -> h1 in B2
    layer(B2, 64,  11);   // c2_*  -> h2 in B2 (hin reads finish before overwrite)
    layer(B2, 128, 19);   // c3_*  -> h3 in B2

    k_head1<<<(N_GRAPHS * 192 + TB - 1) / TB, TB, 0, stream>>>(pools, P(27), P(28), hid);
    k_head2<<<(N_GRAPHS + TB - 1) / TB, TB, 0, stream>>>(hid, P(29), P(30), (float*)d_out);
}